// RerankwMDA_3212635537552
// MI455X (gfx1250) — compile-verified
//
#include <hip/hip_runtime.h>
#include <hip/hip_bf16.h>

// Problem constants (from reference)
constexpr int M_    = 400;
constexpr int KP1   = 10;
constexpr int Q_    = 64;
constexpr int NDB   = 6000;
constexpr int D_    = 1024;

typedef float v2f __attribute__((ext_vector_type(2)));
typedef float v8f __attribute__((ext_vector_type(8)));

// ---------------------------------------------------------------------------
// Kernel 1: X1[q,d] = max_{k<10} x_dba[q, pre[q,k], d]
// ---------------------------------------------------------------------------
__global__ void __launch_bounds__(256) x1_max_kernel(const float* __restrict__ x_dba,
                                                     const int* __restrict__ pre,
                                                     float* __restrict__ X1) {
    int q = blockIdx.x;
    __shared__ int idx[KP1];
    if (threadIdx.x < KP1) idx[threadIdx.x] = pre[q * M_ + threadIdx.x];
    __syncthreads();
    const float* base = x_dba + (size_t)q * NDB * D_;
    for (int d = threadIdx.x; d < D_; d += 256) {
        float mx = base[(size_t)idx[0] * D_ + d];
        #pragma unroll
        for (int k = 1; k < KP1; ++k)
            mx = fmaxf(mx, base[(size_t)idx[k] * D_ + d]);
        X1[(size_t)q * D_ + d] = mx;
    }
}

// ---------------------------------------------------------------------------
// Kernel 2: res_rerank[q,m] = dot(X1[q,:], x_dba[q, pre[q,m], :])
// One wave (32 lanes) per (q, 16-m tile). fp32 WMMA 16x16x4:
//   A (16x4)  : 16 gathered X2 rows x 4 d-values
//     lanes 0-15 : VGPR0=K0, VGPR1=K1 ; lanes 16-31 : VGPR0=K2, VGPR1=K3
//   B (4x16)  : X1 d-chunk replicated across all 16 N-columns (from LDS)
//   C (16x16) : every column identical = the 16 dot products
// ---------------------------------------------------------------------------
__global__ void __launch_bounds__(32) dot_wmma_kernel(const float* __restrict__ x_dba,
                                                      const int* __restrict__ pre,
                                                      const float* __restrict__ X1,
                                                      float* __restrict__ resR) {
    const int q    = blockIdx.y;
    const int m0   = blockIdx.x * 16;
    const int lane = threadIdx.x;
    const int row  = lane & 15;
    const int half = lane >> 4;          // 0: K=0,1   1: K=2,3

    __shared__ float sx[D_];
    const float* X1q = X1 + (size_t)q * D_;
    #pragma unroll
    for (int i = lane; i < D_; i += 32) sx[i] = X1q[i];
    __syncthreads();

    const int idx = pre[q * M_ + m0 + row];
    const float* aBase = x_dba + ((size_t)q * NDB + (size_t)idx) * D_ + half * 2;
    const float* bBase = sx + half * 2;

    v8f acc0 = {}, acc1 = {}, acc2 = {}, acc3 = {};
    for (int d0 = 0; d0 < D_; d0 += 16) {
        float2 a0 = *(const float2*)(aBase + d0);
        float2 a1 = *(const float2*)(aBase + d0 + 4);
        float2 a2 = *(const float2*)(aBase + d0 + 8);
        float2 a3 = *(const float2*)(aBase + d0 + 12);
        float2 b0 = *(const float2*)(bBase + d0);
        float2 b1 = *(const float2*)(bBase + d0 + 4);
        float2 b2 = *(const float2*)(bBase + d0 + 8);
        float2 b3 = *(const float2*)(bBase + d0 + 12);
        v2f A0 = {a0.x, a0.y}, A1 = {a1.x, a1.y}, A2 = {a2.x, a2.y}, A3 = {a3.x, a3.y};
        v2f B0 = {b0.x, b0.y}, B1 = {b1.x, b1.y}, B2 = {b2.x, b2.y}, B3 = {b3.x, b3.y};
        acc0 = __builtin_amdgcn_wmma_f32_16x16x4_f32(false, A0, false, B0, (short)0, acc0, false, false);
        acc1 = __builtin_amdgcn_wmma_f32_16x16x4_f32(false, A1, false, B1, (short)0, acc1, false, false);
        acc2 = __builtin_amdgcn_wmma_f32_16x16x4_f32(false, A2, false, B2, (short)0, acc2, false, false);
        acc3 = __builtin_amdgcn_wmma_f32_16x16x4_f32(false, A3, false, B3, (short)0, acc3, false, false);
    }
    v8f acc = (acc0 + acc1) + (acc2 + acc3);

    // C/D layout: VGPR r, lanes 0-15 -> M=r (N=lane), lanes 16-31 -> M=r+8.
    // Column N==0 lives in lane 0 (M=0..7) and lane 16 (M=8..15).
    if (row == 0) {
        const int mbase = m0 + half * 8;
        #pragma unroll
        for (int r = 0; r < 8; ++r)
            resR[q * M_ + mbase + r] = acc[r];
    }
}

// ---------------------------------------------------------------------------
// Kernel 3: per-query descending sort of res_top1000, combine, stable
// descending argsort of combined scores, scatter rerank_dba_final.
// rank_i = #{j: v_j > v_i} + #{j < i: v_j == v_i}  (exactly stable argsort(-v))
// ---------------------------------------------------------------------------
__global__ void __launch_bounds__(512) sort_scatter_kernel(const float* __restrict__ res_top,
                                                           const float* __restrict__ resR,
                                                           const int* __restrict__ rerank_final,
                                                           int* __restrict__ out) {
    const int q = blockIdx.x;
    const int i = threadIdx.x;
    __shared__ float vals[M_];
    __shared__ float sorted[M_];
    __shared__ float comb[M_];

    if (i < M_) vals[i] = res_top[q * M_ + i];
    __syncthreads();
    if (i < M_) {
        float v = vals[i];
        int cnt = 0;
        for (int j = 0; j < M_; ++j) {
            float vj = vals[j];
            cnt += (vj > v) || (vj == v && j < i);
        }
        sorted[cnt] = v;   // descending value sort
    }
    __syncthreads();
    if (i < M_) comb[i] = (sorted[i] + resR[q * M_ + i]) * 0.5f;
    __syncthreads();
    if (i < M_) {
        float v = comb[i];
        int cnt = 0;
        for (int j = 0; j < M_; ++j) {
            float vj = comb[j];
            cnt += (vj > v) || (vj == v && j < i);
        }
        // out is (N_DB, Q) row-major; reranked occupies rows [0, M)
        out[(size_t)cnt * Q_ + q] = rerank_final[q * M_ + i];
    }
}

// ---------------------------------------------------------------------------
// Kernel 4: tail rows: out[n,q] = ranks[n,q] for n >= M  (flat int4 copy)
// ---------------------------------------------------------------------------
__global__ void tail_copy_kernel(const int* __restrict__ ranks, int* __restrict__ out) {
    const size_t start = (size_t)M_ * Q_;          // 25600 ints, 16B aligned
    const size_t n4 = ((size_t)NDB * Q_ - start) / 4;
    const int4* src = (const int4*)(ranks + start);
    int4* dst = (int4*)(out + start);
    for (size_t t = (size_t)blockIdx.x * blockDim.x + threadIdx.x; t < n4;
         t += (size_t)gridDim.x * blockDim.x)
        dst[t] = src[t];
}

// ---------------------------------------------------------------------------
extern "C" void kernel_launch(void* const* d_in, const int* in_sizes, int n_in,
                              void* d_out, int out_size, void* d_ws, size_t ws_size,
                              hipStream_t stream) {
    const int*   ranks        = (const int*)d_in[0];   // (N_DB, Q) int32
    const int*   rerank_final = (const int*)d_in[1];   // (Q, M)   int32
    const float* res_top      = (const float*)d_in[2]; // (Q, M)   f32
    const int*   pre          = (const int*)d_in[3];   // (Q, M)   int32
    const float* x_dba        = (const float*)d_in[4]; // (Q, N_DB, D) f32
    int* out = (int*)d_out;                            // (N_DB, Q) int32

    float* X1   = (float*)d_ws;                        // Q*D floats   (256 KB)
    float* resR = X1 + (size_t)Q_ * D_;                // Q*M floats   (100 KB)

    x1_max_kernel<<<Q_, 256, 0, stream>>>(x_dba, pre, X1);
    dot_wmma_kernel<<<dim3(M_ / 16, Q_), 32, 0, stream>>>(x_dba, pre, X1, resR);
    sort_scatter_kernel<<<Q_, 512, 0, stream>>>(res_top, resR, rerank_final, out);
    tail_copy_kernel<<<128, 256, 0, stream>>>(ranks, out);
}